// adjust_46368466927635
// MI455X (gfx1250) — compile-verified
//
#include <hip/hip_runtime.h>

// ---------------------------------------------------------------------------
// DCNv2-style fused pipeline for MI455X (gfx1250, wave32, WMMA f32 16x16x4)
//   K1: offset_feat = W_off @ concat(feat_u, 2*feat_s)      (1x1 conv GEMM)
//   K2: om = conv3x3(offset_feat, W_om) + b_om              (implicit GEMM)
//   K3: bilinear deformable sampling -> LDS col -> GEMM with W_dcn, +b, ReLU
// Each wave computes a 16(M) x 32(N) tile: one A fragment feeds two WMMAs.
// ---------------------------------------------------------------------------

typedef float v2f __attribute__((ext_vector_type(2)));
typedef float v8f __attribute__((ext_vector_type(8)));

#define Bn   4
#define Cc   128
#define Gg   4
#define K9   9
#define CgN  32
#define Hh   96
#define Wd_  96
#define HW   (Hh * Wd_)        // 9216
#define KD   (Cc * K9)         // 1152
#define OMC  (3 * Gg * K9)     // 108
#define LDP  33                // padded col row stride (bank-conflict-free)

static __device__ __forceinline__ v8f wmma_f32_k4(v2f a, v2f b, v8f c) {
    // D(16x16,f32) = A(16x4,f32) * B(4x16,f32) + C
    return __builtin_amdgcn_wmma_f32_16x16x4_f32(
        /*neg_a=*/false, a, /*neg_b=*/false, b,
        /*c_mod=*/(short)0, c, /*reuse_a=*/false, /*reuse_b=*/false);
}

// ---------------------------------------------------------------------------
// Kernel 1: offset_feat[b, m, p] = sum_k W_off[m,k] * X[k,p]
//   X rows 0..127 = feat_u, rows 128..255 = 2*feat_s (scale folded into A).
// grid(72, 8, 4), block 128: 4 waves, each a 16x32 tile.
// ---------------------------------------------------------------------------
__global__ __launch_bounds__(128) void k_offset_gemm(
    const float* __restrict__ fu, const float* __restrict__ fs,
    const float* __restrict__ Woff, float* __restrict__ off)
{
    const int lane = threadIdx.x & 31;
    const int wave = threadIdx.x >> 5;
    const int lo = lane & 15, hi = lane >> 4;
    const int n0 = (blockIdx.x * 4 + wave) * 32;
    const int m0 = blockIdx.y * 16;
    const int b  = blockIdx.z;

    const float* pA  = Woff + (size_t)(m0 + lo) * (2 * Cc) + 2 * hi;
    const float* pBu = fu + (size_t)b * Cc * HW + (size_t)(2 * hi) * HW + n0;
    const float* pBs = fs + (size_t)b * Cc * HW + (size_t)(2 * hi) * HW + n0;

    v8f acc0 = {}, acc1 = {};

    #pragma unroll 4
    for (int k = 0; k < Cc; k += 4) {              // feat_u half, scale 1
        v2f a = *(const v2f*)(pA + k);
        const float* s = pBu + (size_t)k * HW;
        v2f b0, b1;
        b0.x = s[lo];      b0.y = s[HW + lo];
        b1.x = s[16 + lo]; b1.y = s[HW + 16 + lo];
        acc0 = wmma_f32_k4(a, b0, acc0);
        acc1 = wmma_f32_k4(a, b1, acc1);
    }
    #pragma unroll 4
    for (int k = 0; k < Cc; k += 4) {              // feat_s half, scale 2 on A
        v2f a = *(const v2f*)(pA + Cc + k);
        a = a * 2.0f;
        const float* s = pBs + (size_t)k * HW;
        v2f b0, b1;
        b0.x = s[lo];      b0.y = s[HW + lo];
        b1.x = s[16 + lo]; b1.y = s[HW + 16 + lo];
        acc0 = wmma_f32_k4(a, b0, acc0);
        acc1 = wmma_f32_k4(a, b1, acc1);
    }

    float* outb = off + ((size_t)b * Cc + m0) * HW;
    #pragma unroll
    for (int r = 0; r < 8; ++r) {
        outb[(size_t)(r + 8 * hi) * HW + n0 + lo]      = acc0[r];
        outb[(size_t)(r + 8 * hi) * HW + n0 + 16 + lo] = acc1[r];
    }
}

// ---------------------------------------------------------------------------
// Kernel 2: om = conv3x3(offset_feat, W_om) + b_om (implicit GEMM, K=1152)
// grid(72, 7, 4), block 128: 4 waves, each a 16x32 tile (M padded 108->112).
// ---------------------------------------------------------------------------
__global__ __launch_bounds__(128) void k_om_conv(
    const float* __restrict__ off, const float* __restrict__ Wom,
    const float* __restrict__ bom, float* __restrict__ om)
{
    const int lane = threadIdx.x & 31;
    const int wave = threadIdx.x >> 5;
    const int lo = lane & 15, hi = lane >> 4;
    const int n0 = (blockIdx.x * 4 + wave) * 32;
    const int m0 = blockIdx.y * 16;
    const int b  = blockIdx.z;

    const int p0 = n0 + lo,       p1 = n0 + 16 + lo;
    const int ya = p0 / Wd_,      xa = p0 % Wd_;
    const int yb = p1 / Wd_,      xb = p1 % Wd_;
    const float* offb = off + (size_t)b * Cc * HW;
    const int arow = (m0 + lo > OMC - 1) ? (OMC - 1) : (m0 + lo);  // clamp pad rows
    const float* pA = Wom + (size_t)arow * KD + 2 * hi;

    v8f acc0 = {}, acc1 = {};
    #pragma unroll 2
    for (int k = 0; k < KD; k += 4) {
        v2f a = *(const v2f*)(pA + k);
        v2f b0, b1;
        #pragma unroll
        for (int j = 0; j < 2; ++j) {
            const int kk = k + 2 * hi + j;
            const int c  = kk / 9, r9 = kk % 9;
            const int ky = r9 / 3 - 1, kx = r9 % 3 - 1;
            const float* cp = offb + (size_t)c * HW;
            float va = 0.0f, vb = 0.0f;
            const int y0 = ya + ky, x0 = xa + kx;
            const int y1 = yb + ky, x1 = xb + kx;
            if (y0 >= 0 && y0 < Hh && x0 >= 0 && x0 < Wd_) va = cp[y0 * Wd_ + x0];
            if (y1 >= 0 && y1 < Hh && x1 >= 0 && x1 < Wd_) vb = cp[y1 * Wd_ + x1];
            b0[j] = va;
            b1[j] = vb;
        }
        acc0 = wmma_f32_k4(a, b0, acc0);
        acc1 = wmma_f32_k4(a, b1, acc1);
    }

    #pragma unroll
    for (int r = 0; r < 8; ++r) {
        const int o = m0 + r + 8 * hi;
        if (o < OMC) {
            float* row = om + ((size_t)b * OMC + o) * HW;
            row[p0] = acc0[r] + bom[o];
            row[p1] = acc1[r] + bom[o];
        }
    }
}

// ---------------------------------------------------------------------------
// Kernel 3: deformable sampling + main conv GEMM + bias + ReLU
//   Block tile: 32 pixels x 128 out-channels. col built in LDS two groups at
//   a time (576 x 32, padded stride 33 -> conflict-free build stores).
// grid(288, 4), block 256 (8 waves, each a 16x32 tile).
// ---------------------------------------------------------------------------
__global__ __launch_bounds__(256) void k_dcn_main(
    const float* __restrict__ fs, const float* __restrict__ om,
    const float* __restrict__ Wdcn, const float* __restrict__ bd,
    float* __restrict__ out)
{
    __shared__ float colS[576 * LDP];                  // 76 KB of 320 KB LDS

    const int tid  = threadIdx.x;
    const int lane = tid & 31;                         // lane == cg in build phase
    const int wave = tid >> 5;                         // 0..7
    const int lo = lane & 15, hi = lane >> 4;
    const int n0 = blockIdx.x * 32;
    const int b  = blockIdx.y;
    const int m0 = wave * 16;

    const float* omb = om + (size_t)b * OMC * HW;
    const float* fsb = fs + (size_t)b * Cc * HW;

    __builtin_prefetch(Wdcn + (size_t)(m0 + lo) * KD, 0, 1);   // global_prefetch_b8

    v8f acc0 = {}, acc1 = {};

    for (int phase = 0; phase < 2; ++phase) {
        // ---- build col chunk for groups {2*phase, 2*phase+1} ----
        // combo = gl*288 + k9*32 + n  (576 combos, 72 per wave, lane = cg)
        #pragma unroll 1
        for (int i = 0; i < 72; ++i) {
            const int combo = wave + 8 * i;
            const int n  = combo & 31;
            const int k9 = (combo >> 5) % 9;
            const int gl = combo / 288;
            const int g  = phase * 2 + gl;
            const int p  = n0 + n;
            const int y  = p / Wd_, x = p % Wd_;
            const int gk = g * 9 + k9;

            const float dy = omb[(size_t)(gk)             * HW + p];
            const float dx = omb[(size_t)(Gg * K9 + gk)   * HW + p];
            const float mz = omb[(size_t)(2 * Gg * K9 + gk) * HW + p];
            const float mask = 1.0f / (1.0f + __expf(-mz));

            const float py = (float)(y + k9 / 3 - 1) + dy;
            const float px = (float)(x + k9 % 3 - 1) + dx;
            const float y0f = floorf(py), x0f = floorf(px);
            const float fy = py - y0f, fx = px - x0f;
            const int y0 = (int)y0f, x0 = (int)x0f;

            const float* ch = fsb + (size_t)(g * CgN + lane) * HW;   // lane = cg
            float v = 0.0f;
            if (y0 >= 0 && y0 < Hh) {
                if (x0 >= 0   && x0 < Wd_)     v += (1.0f-fy)*(1.0f-fx) * ch[y0*Wd_ + x0];
                if (x0+1 >= 0 && x0+1 < Wd_)   v += (1.0f-fy)*fx        * ch[y0*Wd_ + x0+1];
            }
            if (y0+1 >= 0 && y0+1 < Hh) {
                if (x0 >= 0   && x0 < Wd_)     v += fy*(1.0f-fx)        * ch[(y0+1)*Wd_ + x0];
                if (x0+1 >= 0 && x0+1 < Wd_)   v += fy*fx               * ch[(y0+1)*Wd_ + x0+1];
            }
            colS[((gl * CgN + lane) * 9 + k9) * LDP + n] = v * mask;
        }
        __syncthreads();

        // ---- GEMM over this 576-wide K chunk ----
        const float* Arow = Wdcn + (size_t)(m0 + lo) * KD + phase * 576 + 2 * hi;
        #pragma unroll 4
        for (int ks = 0; ks < 576; ks += 4) {
            v2f a = *(const v2f*)(Arow + ks);
            const int kb = (ks + 2 * hi) * LDP;
            v2f b0, b1;
            b0.x = colS[kb + lo];            b0.y = colS[kb + LDP + lo];
            b1.x = colS[kb + 16 + lo];       b1.y = colS[kb + LDP + 16 + lo];
            acc0 = wmma_f32_k4(a, b0, acc0);
            acc1 = wmma_f32_k4(a, b1, acc1);
        }
        __syncthreads();
    }

    // ---- epilogue: bias + ReLU ----
    float* outb = out + ((size_t)b * Cc + m0) * HW;
    #pragma unroll
    for (int r = 0; r < 8; ++r) {
        const int o = m0 + r + 8 * hi;
        const float bias = bd[o];
        outb[(size_t)(r + 8 * hi) * HW + n0 + lo]      = fmaxf(acc0[r] + bias, 0.0f);
        outb[(size_t)(r + 8 * hi) * HW + n0 + 16 + lo] = fmaxf(acc1[r] + bias, 0.0f);
    }
}

// ---------------------------------------------------------------------------
extern "C" void kernel_launch(void* const* d_in, const int* in_sizes, int n_in,
                              void* d_out, int out_size, void* d_ws, size_t ws_size,
                              hipStream_t stream)
{
    (void)in_sizes; (void)n_in; (void)out_size; (void)ws_size;

    const float* fu   = (const float*)d_in[0];   // feat_u  [4,128,96,96]
    const float* fs   = (const float*)d_in[1];   // feat_s  [4,128,96,96]
    const float* Woff = (const float*)d_in[2];   // [128, 256]
    const float* Wom  = (const float*)d_in[3];   // [108, 128, 3, 3]
    const float* bom  = (const float*)d_in[4];   // [108]
    const float* Wdcn = (const float*)d_in[5];   // [128, 128, 3, 3]
    const float* bd   = (const float*)d_in[6];   // [128]

    float* off = (float*)d_ws;                           // [4,128,9216] 18.9 MB
    float* om  = off + (size_t)Bn * Cc * HW;             // [4,108,9216] 15.9 MB
    float* out = (float*)d_out;                          // [4,128,9216]

    k_offset_gemm<<<dim3(72, 8, Bn), 128, 0, stream>>>(fu, fs, Woff, off);
    k_om_conv   <<<dim3(72, 7, Bn), 128, 0, stream>>>(off, Wom, bom, om);
    k_dcn_main  <<<dim3(288, Bn),   256, 0, stream>>>(fs, om, Wdcn, bd, out);
}